// GeneMoEGate_73366631350449
// MI455X (gfx1250) — compile-verified
//
#include <hip/hip_runtime.h>
#include <hip/hip_bf16.h>
#include <math.h>

#define HIDDEN 2048
#define N_EXPERTS 64
#define GENE 512
#define TOK_PER_BLOCK 128
#define THREADS_MAIN 128

typedef float v2f __attribute__((ext_vector_type(2)));
typedef float v8f __attribute__((ext_vector_type(8)));

static __device__ __forceinline__ v8f wmma_f32(v2f a, v2f b, v8f c) {
    // V_WMMA_F32_16X16X4_F32 : D = A(16x4,f32) * B(4x16,f32) + C(16x16,f32)
    return __builtin_amdgcn_wmma_f32_16x16x4_f32(
        /*neg_a=*/false, a, /*neg_b=*/false, b,
        /*c_mod=*/(short)0, c, /*reuse_a=*/false, /*reuse_b=*/false);
}

// ---------------------------------------------------------------------------
// Kernel 1: gene-vector logits (pre-scaled by 2/3) + zero aux accumulators.
// ws layout (floats): [0..255] gbias[b][e], [256..511] cnt[b][e], [512..767] sum[b][e]
// ---------------------------------------------------------------------------
__global__ __launch_bounds__(256) void gate_setup(const float* __restrict__ gv,
                                                  const float* __restrict__ wg,
                                                  float* __restrict__ ws) {
    const int t = threadIdx.x;            // 0..255 -> (b,e)
    const int b = t >> 6;
    const int e = t & 63;
    const float* g = gv + (size_t)b * GENE;
    const float* w = wg + (size_t)e * GENE;
    float d = 0.f;
#pragma unroll 4
    for (int i = 0; i < GENE; ++i) d = fmaf(g[i], w[i], d);
    ws[t]       = (2.f / 3.f) * d;   // gbias: logits = h/3 + 2g/3  (TEMP = 0.5)
    ws[256 + t] = 0.f;               // expert pick counts
    ws[512 + t] = 0.f;               // per-(b,e) score sums
}

// ---------------------------------------------------------------------------
// Kernel 2: main fused gate. 4 waves/block; each wave: 32 tokens x 64 experts.
// ---------------------------------------------------------------------------
__global__ __launch_bounds__(THREADS_MAIN) void gate_main(
    const float* __restrict__ x,        // [tokens, HIDDEN]
    const float* __restrict__ wt,       // [E, HIDDEN]
    const float* __restrict__ gbias,    // [B, E] (pre-scaled)
    float* __restrict__ gcnt,           // [B, E]
    float* __restrict__ gsum,           // [B, E]
    float* __restrict__ out,            // [idx | w | aux]
    int seqLen, int nTokens)
{
    __shared__ float smL[TOK_PER_BLOCK * 65];   // logits, padded stride 65
    __shared__ float smG[N_EXPERTS];
    __shared__ float smCnt[N_EXPERTS];
    __shared__ float smInvZ[TOK_PER_BLOCK];

    const int t    = threadIdx.x;
    const int wave = t >> 5;
    const int lane = t & 31;
    const int half = lane >> 4;          // 0: K={0,1}, 1: K={2,3}
    const int l16  = lane & 15;
    const int tok0 = blockIdx.x * TOK_PER_BLOCK;
    const int bIdx = tok0 / seqLen;      // whole block shares one batch row

    if (t < N_EXPERTS) { smG[t] = gbias[bIdx * N_EXPERTS + t]; smCnt[t] = 0.f; }

    // ---------------- phase 1: WMMA GEMM, 32 tokens x 64 experts per wave ----
    v8f acc[2][4];
#pragma unroll
    for (int mt = 0; mt < 2; ++mt)
#pragma unroll
        for (int nt = 0; nt < 4; ++nt) acc[mt][nt] = (v8f)0.f;

    // A frag (16x4 f32): lane l16 = row M, half selects K pair -> contiguous float2
    const float* aBase0 = x + (size_t)(tok0 + wave * 32 + l16) * HIDDEN + 2 * half;
    const float* aBase1 = aBase0 + (size_t)16 * HIDDEN;
    // B frag (4x16 f32): lane l16 = expert col N, half selects K pair
    const float* bBase  = wt + (size_t)l16 * HIDDEN + 2 * half;

#pragma unroll 2
    for (int k = 0; k < HIDDEN; k += 4) {
        v2f a0 = *(const v2f*)(aBase0 + k);
        v2f a1 = *(const v2f*)(aBase1 + k);
#pragma unroll
        for (int nt = 0; nt < 4; ++nt) {
            v2f bf = *(const v2f*)(bBase + (size_t)nt * 16 * HIDDEN + k);
            acc[0][nt] = wmma_f32(a0, bf, acc[0][nt]);
            acc[1][nt] = wmma_f32(a1, bf, acc[1][nt]);
        }
    }

    // C/D layout: vgpr v, lanes 0-15 -> M=v, lanes 16-31 -> M=v+8; N = l16
#pragma unroll
    for (int mt = 0; mt < 2; ++mt)
#pragma unroll
        for (int nt = 0; nt < 4; ++nt)
#pragma unroll
            for (int v = 0; v < 8; ++v)
                smL[(wave * 32 + mt * 16 + 8 * half + v) * 65 + nt * 16 + l16] =
                    acc[mt][nt][v];

    __syncthreads();

    // ---------------- phase 2: per-token softmax + top-2 (1 thread / token) --
    {
        const int tok = tok0 + t;
        float* row = &smL[t * 65];

        // scale h-logits, add gene bias, find top-2 (ties -> lower index first)
        float v1 = -INFINITY, v2 = -INFINITY;
        int i1 = 0, i2 = 0;
#pragma unroll 4
        for (int e = 0; e < N_EXPERTS; ++e) {
            float l = fmaf(row[e], (1.f / 3.f), smG[e]);
            row[e] = l;
            if (l > v1)      { v2 = v1; i2 = i1; v1 = l; i1 = e; }
            else if (l > v2) { v2 = l;  i2 = e; }
        }
        // stable softmax
        float Z = 0.f;
#pragma unroll 4
        for (int e = 0; e < N_EXPERTS; ++e) {
            float es = __expf(row[e] - v1);
            row[e] = es;
            Z += es;
        }
        const float invZ = 1.f / Z;
        smInvZ[t] = invZ;
        const float s1 = row[i1] * invZ;   // == invZ (exp(0))
        const float s2 = row[i2] * invZ;
        const float dn = 1.f / (s1 + s2 + 1e-20f);

        out[(size_t)tok * 2 + 0]                        = (float)i1;
        out[(size_t)tok * 2 + 1]                        = (float)i2;
        out[(size_t)nTokens * 2 + (size_t)tok * 2 + 0]  = s1 * dn;
        out[(size_t)nTokens * 2 + (size_t)tok * 2 + 1]  = s2 * dn;

        atomicAdd(&smCnt[i1], 1.f);
        atomicAdd(&smCnt[i2], 1.f);
    }

    __syncthreads();

    // ---------------- phase 3: per-expert score sums (transposed, no conflicts)
    if (t < N_EXPERTS) {
        float cs = 0.f;
#pragma unroll 4
        for (int tt = 0; tt < TOK_PER_BLOCK; ++tt)
            cs = fmaf(smL[tt * 65 + t], smInvZ[tt], cs);
        atomicAdd(&gsum[bIdx * N_EXPERTS + t], cs);
        atomicAdd(&gcnt[bIdx * N_EXPERTS + t], smCnt[t]);
    }
}

// ---------------------------------------------------------------------------
// Kernel 3: aux loss = alpha * mean_b sum_e (cnt*E/(s*k)) * (sum/s)
// ---------------------------------------------------------------------------
__global__ __launch_bounds__(256) void gate_finalize(const float* __restrict__ gcnt,
                                                     const float* __restrict__ gsum,
                                                     float* __restrict__ out,
                                                     int seqLen, int bCount, int nTokens) {
    __shared__ float red[256];
    const int t = threadIdx.x;
    red[t] = (t < bCount * N_EXPERTS) ? gcnt[t] * gsum[t] : 0.f;
    __syncthreads();
    for (int s = 128; s > 0; s >>= 1) {
        if (t < s) red[t] += red[t + s];
        __syncthreads();
    }
    if (t == 0) {
        const float scale = 0.001f * (1.f / (float)bCount) *
                            ((float)N_EXPERTS / (2.f * (float)seqLen)) *
                            (1.f / (float)seqLen);
        out[(size_t)nTokens * 4] = red[0] * scale;
    }
}

// ---------------------------------------------------------------------------
extern "C" void kernel_launch(void* const* d_in, const int* in_sizes, int n_in,
                              void* d_out, int out_size, void* d_ws, size_t ws_size,
                              hipStream_t stream) {
    const float* x  = (const float*)d_in[0];   // hidden_states [b,s,H]
    const float* gv = (const float*)d_in[1];   // gene_vectors [b,G]
    const float* wt = (const float*)d_in[2];   // weight_token [E,H]
    const float* wg = (const float*)d_in[3];   // weight_gene  [E,G]
    float* out = (float*)d_out;

    const int nTokens = in_sizes[0] / HIDDEN;      // 16384
    const int bCount  = in_sizes[1] / GENE;        // 4
    const int seqLen  = nTokens / bCount;          // 4096

    float* ws    = (float*)d_ws;
    float* gbias = ws;            // 256
    float* gcnt  = ws + 256;      // 256
    float* gsum  = ws + 512;      // 256

    gate_setup<<<1, 256, 0, stream>>>(gv, wg, ws);
    gate_main<<<nTokens / TOK_PER_BLOCK, THREADS_MAIN, 0, stream>>>(
        x, wt, gbias, gcnt, gsum, out, seqLen, nTokens);
    gate_finalize<<<1, 256, 0, stream>>>(gcnt, gsum, out, seqLen, bCount, nTokens);
}